// TreeCrfLoss_35519379538227
// MI455X (gfx1250) — compile-verified
//
#include <hip/hip_runtime.h>
#include <math.h>

// TreeCRF loss for MI455X (gfx1250, wave32).
// Memory-bound (~70 MB traffic -> ~3us at 23.3 TB/s). Unary term computed as
// trace( sum_tiles U_tile x Onehot_tile ) using chained V_WMMA_F32_16X16X4_F32
// (exact in f32: onehot entries are 0/1). Edge term is a line-granular gather.
// Deterministic two-pass reduction (block partials in d_ws, fixed-order final).

typedef __attribute__((ext_vector_type(2))) float v2f;
typedef __attribute__((ext_vector_type(8))) float v8f;

#define LBL 16
#define NB_UNARY 512
#define NB_EDGE  1024
#define THREADS  256

__device__ __forceinline__ float wave_reduce_add(float v) {
#pragma unroll
  for (int off = 16; off > 0; off >>= 1)
    v += __shfl_xor(v, off, 32);
  return v;
}

__device__ __forceinline__ void block_reduce_store(float v, float* out_slot) {
  __shared__ float red[THREADS / 32];
  const int lane = threadIdx.x & 31;
  const int wave = threadIdx.x >> 5;
  v = wave_reduce_add(v);
  if (lane == 0) red[wave] = v;
  __syncthreads();
  if (wave == 0) {
    const int nw = blockDim.x >> 5;
    float s = (lane < nw) ? red[lane] : 0.0f;
    s = wave_reduce_add(s);
    if (lane == 0) *out_slot = s;
  }
}

// pot_unary = sum_i U[i, tl[i]]  (every node counted exactly once for this
// K-ary tree: internal nodes via the internal term, leaves via the leaf-child
// term; root is internal for N>1).
// Per 16-node tile: D += U_tile(16x16) x O_tile(16x16), O[l][n] = (tl_n == l).
// trace of the accumulated D over all tiles = pot_unary (trace is linear).
__global__ void unary_pot_kernel(const float* __restrict__ U,
                                 const int* __restrict__ tl,
                                 int n_nodes, float* __restrict__ partials) {
  const int lane = threadIdx.x & 31;
  const int wave = threadIdx.x >> 5;
  const int wavesPerBlk = blockDim.x >> 5;
  const int gwave  = blockIdx.x * wavesPerBlk + wave;
  const int nwaves = gridDim.x * wavesPerBlk;
  const int ntiles = n_nodes >> 4;      // full 16-node tiles
  const int h = lane >> 4;              // half-wave: 0 or 1
  const int r = lane & 15;              // node-in-tile == A row == B column

  v8f c = {0.f, 0.f, 0.f, 0.f, 0.f, 0.f, 0.f, 0.f};
  float sacc = 0.0f;

  for (int t = gwave; t < ntiles; t += nwaves) {   // wave-uniform: EXEC all-1s
    const float* row = U + (((size_t)t << 8) + ((size_t)r << 4));
    // A-matrix layout (32-bit 16x4, wave32): VGPR0 = K0|K2, VGPR1 = K1|K3.
    // Chunk k covers labels 4k..4k+3 -> lane needs cols {4k+2h, 4k+2h+1}.
    v2f a[4];
#pragma unroll
    for (int k = 0; k < 4; ++k)
      a[k] = *(const v2f*)(row + 4 * k + 2 * h);   // 8B-aligned
    const int lab = tl[(t << 4) + r];
#pragma unroll
    for (int k = 0; k < 4; ++k) {
      v2f b;                                        // onehot rows of chunk k
      b.x = (lab == 4 * k + 2 * h)     ? 1.0f : 0.0f;
      b.y = (lab == 4 * k + 2 * h + 1) ? 1.0f : 0.0f;
      c = __builtin_amdgcn_wmma_f32_16x16x4_f32(
          /*neg_a=*/false, a[k], /*neg_b=*/false, b,
          /*c_mod=*/(short)0, c, /*reuse_a=*/false, /*reuse_b=*/false);
    }
  }

  // Scalar tail for n_nodes % 16 (empty for N = 500000).
  const int tailBase = ntiles << 4;
  if (gwave == 0 && lane < (n_nodes - tailBase)) {
    const int i = tailBase + lane;
    sacc += U[((size_t)i << 4) + tl[i]];
  }

  // Extract the diagonal of the accumulated C/D matrix once per wave.
  // Element (m,m): m<8 -> VGPR m, lane m; m>=8 -> VGPR m-8, lane m+16.
  float v = sacc;
#pragma unroll
  for (int j = 0; j < 8; ++j) {
    const float dj = c[j];
    if (lane == j || lane == 24 + j) v += dj;
  }

  block_reduce_store(v, &partials[blockIdx.x]);
}

// pot_edge = sum_e edge[child_e, tl[parent_e], tl[child_e]]
__global__ void edge_pot_kernel(const float* __restrict__ E,
                                const int* __restrict__ tl,
                                const int* __restrict__ pidx,
                                const int* __restrict__ cidx,
                                int n_edges, float* __restrict__ partials) {
  float acc = 0.0f;
  for (int e = blockIdx.x * blockDim.x + threadIdx.x; e < n_edges;
       e += gridDim.x * blockDim.x) {
    const int cnode = cidx[e];          // coalesced (== e+1)
    const int pnode = pidx[e];          // coalesced
    const int ci = tl[cnode];           // coalesced gather
    const int pi = tl[pnode];           // 4-way broadcast gather
    acc += E[((size_t)cnode << 8) + ((size_t)pi << 4) + (size_t)ci];
  }
  block_reduce_store(acc, &partials[blockIdx.x]);
}

// Fixed-order reduction of all block partials + logsumexp(beliefs[0]); one wave.
__global__ void final_kernel(const float* __restrict__ ws, int npart,
                             const float* __restrict__ beliefs,
                             float* __restrict__ out) {
  const int lane = threadIdx.x;
  float s = 0.0f;
  for (int i = lane; i < npart; i += 32) s += ws[i];
  s = wave_reduce_add(s);
  if (lane == 0) {
    float m = -INFINITY;
    for (int l = 0; l < LBL; ++l) m = fmaxf(m, beliefs[l]);
    float es = 0.0f;
    for (int l = 0; l < LBL; ++l) es += expf(beliefs[l] - m);
    const float z = m + logf(es);
    out[0] = z - s;                     // -(pot - z)
  }
}

extern "C" void kernel_launch(void* const* d_in, const int* in_sizes, int n_in,
                              void* d_out, int out_size, void* d_ws, size_t ws_size,
                              hipStream_t stream) {
  const float* unary   = (const float*)d_in[0];   // [N,16]
  const float* edge    = (const float*)d_in[1];   // [N,16,16]
  const float* beliefs = (const float*)d_in[2];   // [N,16]
  const int*   tl      = (const int*)d_in[3];     // [N]
  const int*   pidx    = (const int*)d_in[4];     // [N-1]
  const int*   cidx    = (const int*)d_in[5];     // [N-1]
  // d_in[6] internal_mask: unused (unary weights are identically 1, see above)

  const int n_nodes = in_sizes[3];
  const int n_edges = in_sizes[4];

  float* ws = (float*)d_ws;                       // [NB_UNARY + NB_EDGE] floats

  unary_pot_kernel<<<NB_UNARY, THREADS, 0, stream>>>(unary, tl, n_nodes, ws);
  edge_pot_kernel<<<NB_EDGE, THREADS, 0, stream>>>(edge, tl, pidx, cidx,
                                                   n_edges, ws + NB_UNARY);
  final_kernel<<<1, 32, 0, stream>>>(ws, NB_UNARY + NB_EDGE, beliefs,
                                     (float*)d_out);
}